// SAGraphPooling_85452669321524
// MI455X (gfx1250) — compile-verified
//
#include <hip/hip_runtime.h>
#include <hip/hip_bf16.h>

typedef __attribute__((ext_vector_type(2))) float v2f;
typedef __attribute__((ext_vector_type(8))) float v8f;

#define B_ 4
#define N_ 4096
#define F_ 256
#define K_ 2048

// ---------- kernel 1: xw[b,n] = dot(X[b,n,:], w) ----------
// one wave (32 lanes) per row; coalesced strided reads + shuffle reduction
__global__ void xw_kernel(const float* __restrict__ X, const float* __restrict__ w,
                          float* __restrict__ xw) {
  int lane = threadIdx.x & 31;
  int wave = threadIdx.x >> 5;
  int row = blockIdx.x * 8 + wave;  // global row in [0, B*N)
  const float* xr = X + (size_t)row * F_;
  float acc = 0.f;
#pragma unroll
  for (int t = 0; t < F_ / 32; ++t) {
    int f = t * 32 + lane;
    acc += xr[f] * w[f];
  }
#pragma unroll
  for (int off = 16; off > 0; off >>= 1)
    acc += __shfl_xor(acc, off, 32);
  if (lane == 0) xw[row] = acc;
}

// ---------- kernel 2: logits[b,n] = dot(A[b,n,:], xw[b,:]) via f32 WMMA ----------
// one wave per 16 rows. A-operand = xw chunk broadcast across M; B-operand = 16 A rows.
// 16x64 f32 tile staged in LDS (row stride 68 floats -> conflict-free ds_load_b64).
__global__ void logits_wmma_kernel(const float* __restrict__ A, const float* __restrict__ xw,
                                   float* __restrict__ logits) {
  __shared__ float tile[16 * 68];
  __shared__ float vch[64];
  int lane = threadIdx.x;            // 0..31, one full wave
  int b = blockIdx.x >> 8;           // 256 row-groups per batch
  int rowBase = (blockIdx.x & 255) << 4;
  const float* Ab = A + (size_t)b * N_ * N_;
  const float* vb = xw + b * N_;
  int half = lane >> 4;              // 0: lanes 0-15, 1: lanes 16-31
  int lrow = lane & 15;
  v8f c = {0.f, 0.f, 0.f, 0.f, 0.f, 0.f, 0.f, 0.f};

  for (int k0 = 0; k0 < N_; k0 += 64) {
    // coalesced stage of A[rowBase..+15, k0..+63] into padded LDS tile
#pragma unroll
    for (int t = 0; t < 8; ++t) {
      int r = t * 2 + half;
      int col = lrow * 4;
      float4 d = *(const float4*)(Ab + (size_t)(rowBase + r) * N_ + (k0 + col));
      *(float4*)&tile[r * 68 + col] = d;
    }
    vch[lane] = vb[k0 + lane];
    vch[lane + 32] = vb[k0 + lane + 32];
    __syncthreads();

#pragma unroll
    for (int cc = 0; cc < 16; ++cc) {
      int kb = cc * 4 + half * 2;
      // A-matrix 16x4 f32 layout: VGPR0 = K(2*half), VGPR1 = K(2*half+1); broadcast over M
      v2f a;
      a.x = vch[kb];
      a.y = vch[kb + 1];
      // B-matrix 4x16 f32 layout (mirror of A): N = lane%16, VGPR0/1 = K(2*half)/K(2*half+1)
      const float* bp = &tile[lrow * 68 + kb];
      v2f bm;
      bm.x = bp[0];
      bm.y = bp[1];
      c = __builtin_amdgcn_wmma_f32_16x16x4_f32(false, a, false, bm,
                                                (short)0, c, false, false);
    }
    __syncthreads();
  }
  // D layout: VGPR0 lanes 0-15 hold row M=0, N=lane -> logits of rows rowBase+lane
  if (lane < 16) logits[b * N_ + rowBase + lane] = c[0];
}

// ---------- kernel 3: softmax over node axis, per batch ----------
__global__ void softmax_kernel(const float* __restrict__ logits, float* __restrict__ scoring) {
  __shared__ float red[256];
  __shared__ float stat[2];
  int b = blockIdx.x;
  int tid = threadIdx.x;  // 256 threads, 16 elems each
  const float* lb = logits + b * N_;
  float mx = -__builtin_inff();
#pragma unroll
  for (int t = 0; t < 16; ++t) mx = fmaxf(mx, lb[tid + t * 256]);
  red[tid] = mx;
  __syncthreads();
  for (int s = 128; s > 0; s >>= 1) {
    if (tid < s) red[tid] = fmaxf(red[tid], red[tid + s]);
    __syncthreads();
  }
  if (tid == 0) stat[0] = red[0];
  __syncthreads();
  mx = stat[0];
  float sum = 0.f;
#pragma unroll
  for (int t = 0; t < 16; ++t) sum += expf(lb[tid + t * 256] - mx);
  red[tid] = sum;
  __syncthreads();
  for (int s = 128; s > 0; s >>= 1) {
    if (tid < s) red[tid] += red[tid + s];
    __syncthreads();
  }
  if (tid == 0) stat[1] = 1.f / red[0];
  __syncthreads();
  float inv = stat[1];
  float* sb = scoring + b * N_;
#pragma unroll
  for (int t = 0; t < 16; ++t) sb[tid + t * 256] = expf(lb[tid + t * 256] - mx) * inv;
}

// ---------- kernel 4: exact top-K via rank counting (stable, descending) ----------
// position(i) = #{ j : s_j > s_i  or (s_j == s_i and j < i) } -> matches jax.lax.top_k
__global__ void topk_kernel(const float* __restrict__ logits, const float* __restrict__ scoring,
                            int* __restrict__ keep_idx, float* __restrict__ keep_vals) {
  __shared__ float s[N_];
  int b = blockIdx.x >> 4;
  int chunk = blockIdx.x & 15;
  int tid = threadIdx.x;
  const float* lb = logits + b * N_;
#pragma unroll
  for (int t = 0; t < 16; ++t) s[tid + t * 256] = lb[tid + t * 256];
  __syncthreads();
  int i = chunk * 256 + tid;
  float si = s[i];
  int rank = 0;
#pragma unroll 4
  for (int j = 0; j < N_; ++j) {
    float sj = s[j];
    rank += (sj > si) || ((sj == si) && (j < i));
  }
  if (rank < K_) {
    keep_idx[b * K_ + rank] = i;
    keep_vals[b * K_ + rank] = scoring[b * N_ + i];
  }
}

// ---------- kernel 5: Xs_out[b,r,:] = X[b, keep_idx[b,r], :] ----------
__global__ void gatherX_kernel(const float* __restrict__ X, const int* __restrict__ keep_idx,
                               float* __restrict__ Xout) {
  int br = blockIdx.x;  // b*K + r
  int b = br >> 11;
  int src = keep_idx[br];
  Xout[(size_t)br * F_ + threadIdx.x] = X[((size_t)b * N_ + src) * F_ + threadIdx.x];
}

// ---------- kernel 6: As_out[b,r,c] = A[b, idx[r], idx[c]] ----------
// stage full source row + column index list in LDS: global reads coalesced,
// scatter happens only inside LDS, stores coalesced.
__global__ void gatherA_kernel(const float* __restrict__ A, const int* __restrict__ keep_idx,
                               float* __restrict__ Aout) {
  __shared__ float rowbuf[N_];
  __shared__ int cols[K_];
  int br = blockIdx.x;  // b*K + r
  int b = br >> 11;
  int tid = threadIdx.x;
  int srcRow = keep_idx[br];
  const float* rp = A + ((size_t)b * N_ + srcRow) * N_;
#pragma unroll
  for (int t = 0; t < 16; ++t) rowbuf[tid + t * 256] = rp[tid + t * 256];
#pragma unroll
  for (int t = 0; t < 8; ++t) cols[tid + t * 256] = keep_idx[b * K_ + tid + t * 256];
  __syncthreads();
  float* op = Aout + (size_t)br * K_;
#pragma unroll
  for (int t = 0; t < 8; ++t) {
    int cidx = tid + t * 256;
    op[cidx] = rowbuf[cols[cidx]];
  }
}

extern "C" void kernel_launch(void* const* d_in, const int* in_sizes, int n_in,
                              void* d_out, int out_size, void* d_ws, size_t ws_size,
                              hipStream_t stream) {
  const float* Xs = (const float*)d_in[0];  // [B,N,F]
  const float* As = (const float*)d_in[1];  // [B,N,N]
  const float* w  = (const float*)d_in[2];  // [F,1]

  float* out = (float*)d_out;
  float* Xout  = out;                                   // B*K*F floats
  float* Aout  = Xout + (size_t)B_ * K_ * F_;           // B*K*K floats
  float* kvout = Aout + (size_t)B_ * K_ * K_;           // B*K floats

  float* xw      = (float*)d_ws;                        // B*N
  float* logits  = xw + B_ * N_;                        // B*N
  float* scoring = logits + B_ * N_;                    // B*N
  int*   keep_idx = (int*)(scoring + B_ * N_);          // B*K

  xw_kernel<<<B_ * N_ / 8, 256, 0, stream>>>(Xs, w, xw);
  logits_wmma_kernel<<<B_ * (N_ / 16), 32, 0, stream>>>(As, xw, logits);
  softmax_kernel<<<B_, 256, 0, stream>>>(logits, scoring);
  topk_kernel<<<B_ * 16, 256, 0, stream>>>(logits, scoring, keep_idx, kvout);
  gatherX_kernel<<<B_ * K_, F_, 0, stream>>>(Xs, keep_idx, Xout);
  gatherA_kernel<<<B_ * K_, 256, 0, stream>>>(As, keep_idx, Aout);
}